// Linear_37795712205301
// MI455X (gfx1250) — compile-verified
//
#include <hip/hip_runtime.h>

// Embedding gather + masked mean-pool for MI455X (gfx1250).
//
// Shapes (from reference):
//   X:             int32 (16384, 216)   = 16 sparse ids + 4*50 varlen ids
//   sparse_tables: f32   (16, 1000000, 1)
//   varlen_tables: f32   (4, 1000000, 1)
//   out:           f32   (16384, 20)    = [16 gathered | 4 pooled]
//
// Strategy: tables (80 MB total) fit in the 192 MB L2 -> gathers are L2 hits.
// Stream X / out with non-temporal hints so they never evict table lines.

#define B_ROWS     16384
#define N_SPARSE   16
#define N_VARLEN   4
#define L_LEN      50
#define V_SIZE     1000000
#define ROW_STRIDE (N_SPARSE + N_VARLEN * L_LEN)  // 216 ints per row
#define OUT_STRIDE (N_SPARSE + N_VARLEN)          // 20 floats per row
#define EPS_F      1e-8f

typedef int v2i __attribute__((ext_vector_type(2)));

// One thread per (row, sparse_feature). Grid = B*16 threads exactly.
__global__ __launch_bounds__(256) void sparse_gather_kernel(
    const int* __restrict__ X,
    const float* __restrict__ sparse_tables,
    float* __restrict__ out) {
  const int tid = blockIdx.x * blockDim.x + threadIdx.x;  // [0, B*16)
  const int b = tid >> 4;
  const int f = tid & 15;

  // Streaming read of X (used once) -> non-temporal, don't pollute L2.
  const int id = __builtin_nontemporal_load(X + b * ROW_STRIDE + f);

  // Table gather: random but L2-resident (16 * 4 MB tables) -> regular RT load.
  const float e = sparse_tables[f * V_SIZE + id];

  // Streaming write (written once) -> non-temporal store.
  __builtin_nontemporal_store(e, out + b * OUT_STRIDE + f);
}

// One thread per (row, varlen_feature) bag of 50 ids. Grid = B*4 threads.
__global__ __launch_bounds__(256) void varlen_pool_kernel(
    const int* __restrict__ X,
    const float* __restrict__ varlen_tables,
    float* __restrict__ out) {
  const int tid = blockIdx.x * blockDim.x + threadIdx.x;  // [0, B*4)
  const int b = tid >> 2;
  const int v = tid & 3;

  // 50 contiguous ids, 200 bytes, 8-byte aligned for every (b, v):
  //   byte offset = b*864 + 64 + v*200, all multiples of 8.
  const v2i* __restrict__ ids2 = reinterpret_cast<const v2i*>(
      X + b * ROW_STRIDE + N_SPARSE + v * L_LEN);
  const float* __restrict__ tab = varlen_tables + v * V_SIZE;

  float sum = 0.0f;
  float cnt = 0.0f;

  // Fully unrolled: ~50 independent L2 gathers in flight per lane.
  // Branchless mask (id==0 prob ~1e-6): matches reference emb*mask semantics.
#pragma unroll
  for (int i = 0; i < L_LEN / 2; ++i) {
    const v2i p = __builtin_nontemporal_load(ids2 + i);

    const float e0 = tab[p.x];
    const float m0 = (p.x != 0) ? 1.0f : 0.0f;
    sum += e0 * m0;
    cnt += m0;

    const float e1 = tab[p.y];
    const float m1 = (p.y != 0) ? 1.0f : 0.0f;
    sum += e1 * m1;
    cnt += m1;
  }

  const float pooled = sum / (cnt + EPS_F);
  __builtin_nontemporal_store(pooled, out + b * OUT_STRIDE + N_SPARSE + v);
}

extern "C" void kernel_launch(void* const* d_in, const int* in_sizes, int n_in,
                              void* d_out, int out_size, void* d_ws, size_t ws_size,
                              hipStream_t stream) {
  (void)in_sizes; (void)n_in; (void)out_size; (void)d_ws; (void)ws_size;

  const int*   X             = (const int*)d_in[0];    // (B, 216) int32
  const float* sparse_tables = (const float*)d_in[1];  // (16, V) f32
  const float* varlen_tables = (const float*)d_in[2];  // (4, V)  f32
  float*       out           = (float*)d_out;          // (B, 20) f32

  // B*16 = 262144 threads -> 1024 blocks of 256 (exact, no tail).
  sparse_gather_kernel<<<(B_ROWS * N_SPARSE) / 256, 256, 0, stream>>>(
      X, sparse_tables, out);

  // B*4 = 65536 threads -> 256 blocks of 256 (exact, no tail).
  varlen_pool_kernel<<<(B_ROWS * N_VARLEN) / 256, 256, 0, stream>>>(
      X, varlen_tables, out);
}